// SpikingSleepWake_67018669686955
// MI455X (gfx1250) — compile-verified
//
#include <hip/hip_runtime.h>
#include <hip/hip_bf16.h>
#include <stdint.h>

#define NN       4096
#define STEPS    100
#define TC       4                 // timesteps per async chunk
#define NCHUNK   (STEPS / TC)      // 25
#define BLK      256
#define NBLK     (NN / BLK)        // 16

// ---------------------------------------------------------------------------
// Izhikevich RS update (matches reference _izh_step exactly)
// ---------------------------------------------------------------------------
__device__ __forceinline__ void izh_step(float& v, float& u, float& r, float I) {
    float vn  = v + (0.04f * v * v + 5.0f * v + 140.0f - u + I);
    float un  = u + 0.02f * (0.2f * v - u);
    bool  spk = (vn >= 30.0f);
    v = spk ? -65.0f : vn;
    u = un + (spk ? 8.0f : 0.0f);
    r = r * 0.95f + (spk ? 0.05f : 0.0f);
}

// ---------------------------------------------------------------------------
// CDNA5 async global -> LDS (ASYNCcnt-tracked). GV addressing mode:
//   VDST = VGPR holding LDS byte address, VADDR = 64-bit global address.
// ---------------------------------------------------------------------------
__device__ __forceinline__ void async_b128_to_lds(uint32_t lds_addr, const float* gaddr) {
    asm volatile("global_load_async_to_lds_b128 %0, %1, off"
                 :: "v"(lds_addr), "v"(gaddr)
                 : "memory");
}
__device__ __forceinline__ void wait_async_le2() {
    asm volatile("s_wait_asynccnt 0x2" ::: "memory");
}
__device__ __forceinline__ void wait_async_le0() {
    asm volatile("s_wait_asynccnt 0x0" ::: "memory");
}

// ---------------------------------------------------------------------------
// Kernel A: reduce input rate buffers, compute sp / I_wake / I_sleep scalars.
// ---------------------------------------------------------------------------
__global__ __launch_bounds__(BLK) void sw_scalars_kernel(
        const float* __restrict__ wake_rate, const float* __restrict__ sleep_rate,
        const float* __restrict__ mel_p, const float* __restrict__ ar_p,
        const float* __restrict__ th_p,  const float* __restrict__ sp_p,
        float* __restrict__ ws, float* __restrict__ out) {
    __shared__ float red[2][BLK];
    int t = threadIdx.x;
    float sw = 0.0f, ss = 0.0f;
    for (int k = t; k < NN; k += BLK) { sw += wake_rate[k]; ss += sleep_rate[k]; }
    red[0][t] = sw; red[1][t] = ss;
    __syncthreads();
    for (int off = BLK / 2; off > 0; off >>= 1) {
        if (t < off) { red[0][t] += red[0][t + off]; red[1][t] += red[1][t + off]; }
        __syncthreads();
    }
    if (t == 0) {
        float mel = mel_p[0], ar = ar_p[0], th = th_p[0];
        float sp = fminf(fmaxf(sp_p[0] + 0.001f, 0.0f), 1.0f);
        float sleep_drive = mel * 8.0f + sp * 6.0f - ar * 5.0f - th * 10.0f;
        float wake_drive  = ar * 5.0f + th * 10.0f + (1.0f - mel) * 4.0f - sp * 3.0f;
        float mean_wr = red[0][0] * (1.0f / (float)NN);
        float mean_sr = red[1][0] * (1.0f / (float)NN);
        ws[0] = fmaxf(0.0f, wake_drive)  + 3.0f - mean_sr * 8.0f;   // I_wake
        ws[1] = fmaxf(0.0f, sleep_drive) - 2.0f - mean_wr * 8.0f;   // I_sleep
        out[2 * NN + 3] = sp;                                       // sp output
    }
}

// ---------------------------------------------------------------------------
// Kernel B: 100-step scan with double-buffered async noise prefetch into LDS.
// ---------------------------------------------------------------------------
__global__ __launch_bounds__(BLK) void sw_izh_kernel(
        const float* __restrict__ wv_in, const float* __restrict__ wu_in,
        const float* __restrict__ wr_in, const float* __restrict__ sv_in,
        const float* __restrict__ su_in, const float* __restrict__ sr_in,
        const float* __restrict__ wnoise, const float* __restrict__ snoise,
        const float* __restrict__ ws_scalars,
        float* __restrict__ part_w, float* __restrict__ part_s,
        float* __restrict__ out) {
    // [buffer][pop][step*256 + neuron]
    __shared__ float lds[2][2][TC * BLK];
    __shared__ float red[2][BLK];

    const int tid = threadIdx.x;
    const int i0  = blockIdx.x * BLK;
    const int gid = i0 + tid;

    const float I_wake  = ws_scalars[0];
    const float I_sleep = ws_scalars[1];

    float wv = wv_in[gid], wu = wu_in[gid], wr = wr_in[gid];
    float sv = sv_in[gid], su = su_in[gid], sr = sr_in[gid];

    // Each thread issues one b128 per population per chunk:
    //   step-in-chunk = tid>>6, 4 consecutive neurons at (tid&63)*4.
    const int srow = tid >> 6;
    const int scol = (tid & 63) << 2;

    // prologue: chunk 0 into buffer 0
    {
        const float* gw = wnoise + (size_t)srow * NN + i0 + scol;
        const float* gs = snoise + (size_t)srow * NN + i0 + scol;
        async_b128_to_lds((uint32_t)(uintptr_t)&lds[0][0][srow * BLK + scol], gw);
        async_b128_to_lds((uint32_t)(uintptr_t)&lds[0][1][srow * BLK + scol], gs);
    }

    for (int c = 0; c < NCHUNK; ++c) {
        if (c + 1 < NCHUNK) {
            const int nb = (c + 1) & 1;
            const size_t trow = (size_t)((c + 1) * TC + srow) * NN + i0 + scol;
            async_b128_to_lds((uint32_t)(uintptr_t)&lds[nb][0][srow * BLK + scol], wnoise + trow);
            async_b128_to_lds((uint32_t)(uintptr_t)&lds[nb][1][srow * BLK + scol], snoise + trow);
            wait_async_le2();   // chunk c's 2 loads (in-order) are complete
        } else {
            wait_async_le0();
        }
        __syncthreads();        // chunk c visible to all waves

        const int b = c & 1;
        #pragma unroll
        for (int s = 0; s < TC; ++s) {
            float nw = lds[b][0][s * BLK + tid];
            float ns = lds[b][1][s * BLK + tid];
            izh_step(wv, wu, wr, I_wake  + nw * 0.3f);
            izh_step(sv, su, sr, I_sleep + ns * 0.3f);
        }
        __syncthreads();        // safe to overwrite this buffer next round
    }

    out[gid]      = wr;
    out[NN + gid] = sr;

    // deterministic per-block partial sums for the post-loop means
    red[0][tid] = wr; red[1][tid] = sr;
    __syncthreads();
    for (int off = BLK / 2; off > 0; off >>= 1) {
        if (tid < off) { red[0][tid] += red[0][tid + off]; red[1][tid] += red[1][tid + off]; }
        __syncthreads();
    }
    if (tid == 0) { part_w[blockIdx.x] = red[0][0]; part_s[blockIdx.x] = red[1][0]; }
}

// ---------------------------------------------------------------------------
// Kernel C: fold 16 partials -> means / is_sleeping / responsiveness.
// ---------------------------------------------------------------------------
__global__ void sw_finish_kernel(const float* __restrict__ th_p,
                                 const float* __restrict__ part_w,
                                 const float* __restrict__ part_s,
                                 float* __restrict__ out) {
    if (threadIdx.x == 0 && blockIdx.x == 0) {
        float sw = 0.0f, ss = 0.0f;
        for (int b = 0; b < NBLK; ++b) { sw += part_w[b]; ss += part_s[b]; }
        float wm = sw * (1.0f / (float)NN);
        float sm = ss * (1.0f / (float)NN);
        bool slp = (sm > wm) && (th_p[0] < 0.2f);
        out[2 * NN + 0] = wm;
        out[2 * NN + 1] = sm;
        out[2 * NN + 2] = slp ? 1.0f : 0.0f;
        out[2 * NN + 4] = slp ? 0.3f : 1.0f;
    }
}

// ---------------------------------------------------------------------------
extern "C" void kernel_launch(void* const* d_in, const int* in_sizes, int n_in,
                              void* d_out, int out_size, void* d_ws, size_t ws_size,
                              hipStream_t stream) {
    const float* wake_v     = (const float*)d_in[0];
    const float* wake_u     = (const float*)d_in[1];
    const float* wake_rate  = (const float*)d_in[2];
    const float* sleep_v    = (const float*)d_in[3];
    const float* sleep_u    = (const float*)d_in[4];
    const float* sleep_rate = (const float*)d_in[5];
    const float* wake_noise = (const float*)d_in[6];
    const float* sleep_noise= (const float*)d_in[7];
    const float* mel        = (const float*)d_in[8];
    const float* arousal    = (const float*)d_in[9];
    const float* threat     = (const float*)d_in[10];
    const float* sleep_pres = (const float*)d_in[11];

    float* out    = (float*)d_out;
    float* ws     = (float*)d_ws;       // [0..1]=I_wake/I_sleep, [2..17]=wake partials, [18..33]=sleep partials
    float* part_w = ws + 2;
    float* part_s = ws + 2 + NBLK;

    sw_scalars_kernel<<<1, BLK, 0, stream>>>(wake_rate, sleep_rate, mel, arousal,
                                             threat, sleep_pres, ws, out);
    sw_izh_kernel<<<NBLK, BLK, 0, stream>>>(wake_v, wake_u, wake_rate,
                                            sleep_v, sleep_u, sleep_rate,
                                            wake_noise, sleep_noise,
                                            ws, part_w, part_s, out);
    sw_finish_kernel<<<1, 32, 0, stream>>>(threat, part_w, part_s, out);
}